// GCNLinkPredictor_43954695308083
// MI455X (gfx1250) — compile-verified
//
#include <hip/hip_runtime.h>
#include <hip/hip_bf16.h>

typedef float v2f __attribute__((ext_vector_type(2)));
typedef float v8f __attribute__((ext_vector_type(8)));

#define DIM 128

// ---------------------------------------------------------------------------
// deg[dst[e]] += 1
// ---------------------------------------------------------------------------
__global__ void deg_kernel(const int* __restrict__ dst, float* __restrict__ deg, int E) {
    int e = blockIdx.x * blockDim.x + threadIdx.x;
    if (e < E) atomicAdd(&deg[dst[e]], 1.0f);
}

// deg -> dinv in place: d>0 ? 1/sqrt(d) : 0
__global__ void dinv_kernel(float* __restrict__ deg, int N) {
    int i = blockIdx.x * blockDim.x + threadIdx.x;
    if (i < N) {
        float d = deg[i];
        deg[i] = (d > 0.0f) ? rsqrtf(d) : 0.0f;
    }
}

// ---------------------------------------------------------------------------
// H[N x 128] = X[N x 128] @ W[128 x 128] using V_WMMA_F32_16X16X4_F32.
// Block = (32, 8): 8 waves, wave y computes the 16x16 tile at columns y*16.
// Grid.x = N/16 (N divisible by 16 here: 100000 = 6250*16).
//
// ISA lane layout for 32-bit 16x4 A (and 4x16 B), wave32:
//   VGPR0: lanes 0-15 -> K=0, lanes 16-31 -> K=2
//   VGPR1: lanes 0-15 -> K=1, lanes 16-31 -> K=3
// C/D 16x16: VGPR v, lanes 0-15 -> M=v, lanes 16-31 -> M=v+8; N = lane&15.
// ---------------------------------------------------------------------------
__global__ void gemm128_wmma(const float* __restrict__ X, const float* __restrict__ W,
                             float* __restrict__ H) {
    const int lane = threadIdx.x;          // 0..31
    const int half = lane >> 4;            // 0 | 1
    const int m    = lane & 15;            // row-in-tile for A, col-in-tile for B
    const int rowBase = blockIdx.x * 16;
    const int colBase = threadIdx.y * 16;

    const float* __restrict__ xrow = X + (size_t)(rowBase + m) * DIM;

    v8f c = {};
    #pragma unroll
    for (int k0 = 0; k0 < DIM; k0 += 4) {
        const int ka = k0 + 2 * half;
        v2f a, b;
        a.x = xrow[ka];
        a.y = xrow[ka + 1];
        b.x = W[(size_t)ka * DIM + colBase + m];
        b.y = W[(size_t)(ka + 1) * DIM + colBase + m];
        c = __builtin_amdgcn_wmma_f32_16x16x4_f32(
                /*neg_a=*/false, a, /*neg_b=*/false, b,
                /*c_mod=*/(short)0, c, /*reuse_a=*/false, /*reuse_b=*/false);
    }

    float* __restrict__ hout = H + (size_t)(rowBase + 8 * half) * DIM + colBase + m;
    #pragma unroll
    for (int v = 0; v < 8; ++v) {
        hout[(size_t)v * DIM] = c[v];
    }
}

// ---------------------------------------------------------------------------
// out[dst[e]] += h[src[e]] * dinv[src[e]]*dinv[dst[e]]
// Block (32,8): 8 edges per block, lane x handles 4 consecutive floats.
// ---------------------------------------------------------------------------
__global__ void scatter_edges(const float* __restrict__ h, const int* __restrict__ src,
                              const int* __restrict__ dst, const float* __restrict__ dinv,
                              float* __restrict__ out, int E) {
    int e = blockIdx.x * 8 + threadIdx.y;
    if (e >= E) return;
    const int s = src[e];
    const int d = dst[e];
    const float nrm = dinv[s] * dinv[d];
    const int c = threadIdx.x * 4;
    const float4 v = *(const float4*)(h + (size_t)s * DIM + c);
    float* o = out + (size_t)d * DIM + c;
    atomicAdd(o + 0, v.x * nrm);
    atomicAdd(o + 1, v.y * nrm);
    atomicAdd(o + 2, v.z * nrm);
    atomicAdd(o + 3, v.w * nrm);
}

// ---------------------------------------------------------------------------
// buf[i] = act(buf[i] + b[i % 128]); act = relu if do_relu
// ---------------------------------------------------------------------------
__global__ void bias_act(float* __restrict__ buf, const float* __restrict__ b,
                         int total, int do_relu) {
    int i = blockIdx.x * blockDim.x + threadIdx.x;
    if (i >= total) return;
    float v = buf[i] + b[i & (DIM - 1)];
    if (do_relu) v = fmaxf(v, 0.0f);
    buf[i] = v;
}

// ---------------------------------------------------------------------------
// scores[e] = z[head]^T rel_W[rel] z[tail]
// 128 threads per triple; thread j owns column j (coalesced rel_W reads):
//   s = sum_j zt[j] * sum_i zh[i]*M[i][j]
// ---------------------------------------------------------------------------
__global__ void score_kernel(const float* __restrict__ z, const float* __restrict__ relW,
                             const int* __restrict__ rel, const int* __restrict__ head,
                             const int* __restrict__ tail, float* __restrict__ out, int ES) {
    const int e = blockIdx.x;
    if (e >= ES) return;
    __shared__ float zh[DIM];
    __shared__ float red[4];

    const int j = threadIdx.x;          // 0..127
    const int hI = head[e];
    const int tI = tail[e];
    const int r  = rel[e];

    zh[j] = z[(size_t)hI * DIM + j];
    __syncthreads();

    const float* __restrict__ M = relW + (size_t)r * DIM * DIM;
    float acc = 0.0f;
    #pragma unroll 8
    for (int i = 0; i < DIM; ++i) {
        acc = fmaf(zh[i], M[(size_t)i * DIM + j], acc);
    }
    acc *= z[(size_t)tI * DIM + j];

    // wave32 reduction, then across the 4 waves via LDS
    #pragma unroll
    for (int off = 16; off > 0; off >>= 1)
        acc += __shfl_down(acc, off, 32);
    if ((j & 31) == 0) red[j >> 5] = acc;
    __syncthreads();
    if (j == 0) out[e] = red[0] + red[1] + red[2] + red[3];
}

// ---------------------------------------------------------------------------
extern "C" void kernel_launch(void* const* d_in, const int* in_sizes, int n_in,
                              void* d_out, int out_size, void* d_ws, size_t ws_size,
                              hipStream_t stream) {
    const float* x0    = (const float*)d_in[0];
    const float* W1    = (const float*)d_in[1];
    const float* b1    = (const float*)d_in[2];
    const float* W2    = (const float*)d_in[3];
    const float* b2    = (const float*)d_in[4];
    const float* relW  = (const float*)d_in[5];
    const int*   eidx  = (const int*)d_in[6];
    const int*   rel   = (const int*)d_in[7];
    const int*   head  = (const int*)d_in[8];
    const int*   tail  = (const int*)d_in[9];
    float*       out   = (float*)d_out;

    const int N  = in_sizes[0] / DIM;       // 100000
    const int E  = in_sizes[6] / 2;         // 1.6M
    const int ES = in_sizes[7];             // 30000

    const int* src = eidx;
    const int* dst = eidx + E;

    // Workspace layout: deg/dinv (N) | bufA (N*128) | bufB (N*128)
    float* deg  = (float*)d_ws;
    float* bufA = deg + N;
    float* bufB = bufA + (size_t)N * DIM;
    const size_t featBytes = (size_t)N * DIM * sizeof(float);

    // degrees -> dinv (in place)
    hipMemsetAsync(deg, 0, (size_t)N * sizeof(float), stream);
    deg_kernel<<<(E + 255) / 256, 256, 0, stream>>>(dst, deg, E);
    dinv_kernel<<<(N + 255) / 256, 256, 0, stream>>>(deg, N);

    // conv1: h1 = x0 @ W1 ; agg = scatter(norm*h1[src] -> dst) ; x = relu(agg + b1)
    gemm128_wmma<<<N / 16, dim3(32, 8), 0, stream>>>(x0, W1, bufA);
    hipMemsetAsync(bufB, 0, featBytes, stream);
    scatter_edges<<<(E + 7) / 8, dim3(32, 8), 0, stream>>>(bufA, src, dst, deg, bufB, E);
    bias_act<<<(N * DIM + 255) / 256, 256, 0, stream>>>(bufB, b1, N * DIM, 1);

    // conv2: h2 = x @ W2 ; z = scatter(...) + b2
    gemm128_wmma<<<N / 16, dim3(32, 8), 0, stream>>>(bufB, W2, bufA);
    hipMemsetAsync(bufB, 0, featBytes, stream);
    scatter_edges<<<(E + 7) / 8, dim3(32, 8), 0, stream>>>(bufA, src, dst, deg, bufB, E);
    bias_act<<<(N * DIM + 255) / 256, 256, 0, stream>>>(bufB, b2, N * DIM, 0);

    // scores
    score_kernel<<<ES, DIM, 0, stream>>>(bufB, relW, rel, head, tail, out, ES);
}